// Encoder_16492674417012
// MI455X (gfx1250) — compile-verified
//
#include <hip/hip_runtime.h>
#include <hip/hip_bf16.h>

typedef float v2f __attribute__((ext_vector_type(2)));
typedef float v8f __attribute__((ext_vector_type(8)));

#define N_FEAT 128
#define OUT_DIM 64

// ---------------------------------------------------------------------------
// GEMM: Y[m, 0:128] = (X[m] (+ X2[m])) @ W^T + bias      W is [128][128] row-major
// Block = 128 threads (4 waves). Each wave owns 16 rows; block owns 64 rows.
// A/B fragments for V_WMMA_F32_16X16X4_F32:
//   A (16x4):  lanes 0-15 row M=lane, K={k,k+1}; lanes 16-31 row M=lane-16, K={k+2,k+3}
//   B (4x16):  lanes 0-15 col N=lane, K={k,k+1}; lanes 16-31 col N=lane-16, K={k+2,k+3}
//   (B[k][j] = W[j][k], so both read the same [row/col][k..k+1] pattern)
// LDS strides 132 (= 4 mod 64): bank-conflict-free, 8B-aligned pairs.
// ---------------------------------------------------------------------------
__global__ __launch_bounds__(128) void gin_gemm128(
    const float* __restrict__ X, const float* __restrict__ X2,
    const float* __restrict__ W, const float* __restrict__ bias,
    float* __restrict__ Y, int nRows)
{
    __shared__ float sW[128 * 132];
    __shared__ float sX[4][16 * 132];
    __shared__ float sB[128];

    const int tid  = threadIdx.x;
    const int wave = tid >> 5;
    const int lane = tid & 31;

    // Stage W (128x128) into LDS, coalesced float4.
    for (int i = tid; i < 128 * 32; i += 128) {
        const int r  = i >> 5;
        const int c4 = (i & 31) * 4;
        const float4 w = *(const float4*)(W + r * N_FEAT + c4);
        float* d = &sW[r * 132 + c4];
        d[0] = w.x; d[1] = w.y; d[2] = w.z; d[3] = w.w;
    }
    sB[tid] = bias[tid];

    // Stage this wave's 16-row X tile (optionally fused add of X2).
    const long m0 = (long)blockIdx.x * 64 + wave * 16;
    const bool hasX2 = (X2 != nullptr);
    for (int i = lane; i < 16 * 32; i += 32) {
        const int r  = i >> 5;
        const int c4 = (i & 31) * 4;
        long row = m0 + r; if (row >= nRows) row = nRows - 1;   // clamp, stores guarded
        float4 v = *(const float4*)(X + row * N_FEAT + c4);
        if (hasX2) {
            const float4 u = *(const float4*)(X2 + row * N_FEAT + c4);
            v.x += u.x; v.y += u.y; v.z += u.z; v.w += u.w;
        }
        float* d = &sX[wave][r * 132 + c4];
        d[0] = v.x; d[1] = v.y; d[2] = v.z; d[3] = v.w;
    }
    __syncthreads();

    v8f acc[8] = {};
    const int rowA = lane & 15;
    const int koff = (lane >> 4) * 2;          // 0 for lanes 0-15, 2 for 16-31

    for (int kt = 0; kt < 32; ++kt) {
        const int k = kt * 4 + koff;
        const v2f a = *(const v2f*)&sX[wave][rowA * 132 + k];
        #pragma unroll
        for (int jt = 0; jt < 8; ++jt) {
            const v2f b = *(const v2f*)&sW[(jt * 16 + rowA) * 132 + k];
            acc[jt] = __builtin_amdgcn_wmma_f32_16x16x4_f32(
                false, a, false, b, (short)0, acc[jt], false, false);
        }
    }

    // C layout: vgpr r -> row m0+r (lanes 0-15) / m0+8+r (lanes 16-31), col = lane&15 + 16*jt
    const int col0   = lane & 15;
    const long mbase = m0 + ((lane >> 4) ? 8 : 0);
    #pragma unroll
    for (int jt = 0; jt < 8; ++jt) {
        const float bb = sB[jt * 16 + col0];
        #pragma unroll
        for (int r = 0; r < 8; ++r) {
            const long row = mbase + r;
            if (row < nRows)
                Y[row * N_FEAT + jt * 16 + col0] = acc[jt][r] + bb;
        }
    }
}

// ---------------------------------------------------------------------------
// agg[dst] += h[src]  : one 32-lane chunk (float4/lane = 128 feats) per edge
// ---------------------------------------------------------------------------
__global__ __launch_bounds__(256) void gin_scatter(
    const float* __restrict__ H, const int* __restrict__ src,
    const int* __restrict__ dst, float* __restrict__ agg, int nEdges)
{
    const int t  = blockIdx.x * 256 + threadIdx.x;
    const int e  = t >> 5;
    const int c4 = (t & 31) * 4;
    if (e >= nEdges) return;
    const long s = src[e];
    const long d = dst[e];
    const float4 v = *(const float4*)(H + s * N_FEAT + c4);
    float* p = agg + d * N_FEAT + c4;
    atomicAdd(p + 0, v.x);
    atomicAdd(p + 1, v.y);
    atomicAdd(p + 2, v.z);
    atomicAdd(p + 3, v.w);
}

__global__ __launch_bounds__(256) void gin_zero(float* __restrict__ p, long n)
{
    const long i = ((long)blockIdx.x * 256 + threadIdx.x) * 4;
    if (i < n) *(float4*)(p + i) = make_float4(0.f, 0.f, 0.f, 0.f);
}

// ---------------------------------------------------------------------------
// out = softmax( cat(H0..H3) @ Wout^T + bout ), Wout is [64][512] row-major.
// Block = 64 threads (2 waves), 32 rows per block. K = 4 planes x 128.
// Softmax done in-register with __shfl_xor butterflies (stay in 16-lane halves).
// ---------------------------------------------------------------------------
__global__ __launch_bounds__(64) void gin_out_softmax(
    const float* __restrict__ H, const float* __restrict__ Wout,
    const float* __restrict__ bout, float* __restrict__ out,
    int nRows, long planeStride)
{
    __shared__ float sW[64 * 516];
    __shared__ float sX[2][16 * 132];
    __shared__ float sB[64];

    const int tid  = threadIdx.x;
    const int wave = tid >> 5;
    const int lane = tid & 31;

    // Stage Wout (64x512) into LDS (stride 516 = 4 mod 64: conflict-free, 8B aligned).
    for (int i = tid; i < 64 * 128; i += 64) {
        const int r  = i >> 7;
        const int c4 = (i & 127) * 4;
        const float4 w = *(const float4*)(Wout + r * 512 + c4);
        float* d = &sW[r * 516 + c4];
        d[0] = w.x; d[1] = w.y; d[2] = w.z; d[3] = w.w;
    }
    sB[tid] = bout[tid];

    const long m0 = (long)blockIdx.x * 32 + wave * 16;
    v8f acc[4] = {};
    const int rowA = lane & 15;
    const int koff = (lane >> 4) * 2;

    for (int plane = 0; plane < 4; ++plane) {
        __syncthreads();
        const float* P = H + plane * planeStride;
        for (int i = lane; i < 16 * 32; i += 32) {
            const int r  = i >> 5;
            const int c4 = (i & 31) * 4;
            long row = m0 + r; if (row >= nRows) row = nRows - 1;
            const float4 v = *(const float4*)(P + row * N_FEAT + c4);
            float* d = &sX[wave][r * 132 + c4];
            d[0] = v.x; d[1] = v.y; d[2] = v.z; d[3] = v.w;
        }
        for (int kt = 0; kt < 32; ++kt) {
            const int k  = kt * 4 + koff;
            const int kg = plane * 128 + k;
            const v2f a = *(const v2f*)&sX[wave][rowA * 132 + k];
            #pragma unroll
            for (int jt = 0; jt < 4; ++jt) {
                const v2f b = *(const v2f*)&sW[(jt * 16 + rowA) * 516 + kg];
                acc[jt] = __builtin_amdgcn_wmma_f32_16x16x4_f32(
                    false, a, false, b, (short)0, acc[jt], false, false);
            }
        }
    }

    // Bias + row softmax. Row r lives in lanes 0-15 (and row r+8 in lanes 16-31)
    // across the 4 jt tiles -> reduce over 4 regs then xor-shuffle over 16 lanes.
    const int col0   = lane & 15;
    const long mbase = m0 + ((lane >> 4) ? 8 : 0);
    #pragma unroll
    for (int r = 0; r < 8; ++r) {
        float v[4];
        #pragma unroll
        for (int jt = 0; jt < 4; ++jt) v[jt] = acc[jt][r] + sB[jt * 16 + col0];
        float mx = fmaxf(fmaxf(v[0], v[1]), fmaxf(v[2], v[3]));
        mx = fmaxf(mx, __shfl_xor(mx, 1, 32));
        mx = fmaxf(mx, __shfl_xor(mx, 2, 32));
        mx = fmaxf(mx, __shfl_xor(mx, 4, 32));
        mx = fmaxf(mx, __shfl_xor(mx, 8, 32));
        float e[4], sum = 0.f;
        #pragma unroll
        for (int jt = 0; jt < 4; ++jt) { e[jt] = expf(v[jt] - mx); sum += e[jt]; }
        sum += __shfl_xor(sum, 1, 32);
        sum += __shfl_xor(sum, 2, 32);
        sum += __shfl_xor(sum, 4, 32);
        sum += __shfl_xor(sum, 8, 32);
        const float inv = 1.f / sum;
        const long row = mbase + r;
        if (row < nRows) {
            #pragma unroll
            for (int jt = 0; jt < 4; ++jt)
                out[row * OUT_DIM + jt * 16 + col0] = e[jt] * inv;
        }
    }
}

// ---------------------------------------------------------------------------
extern "C" void kernel_launch(void* const* d_in, const int* in_sizes, int n_in,
                              void* d_out, int out_size, void* d_ws, size_t ws_size,
                              hipStream_t stream)
{
    const float* x       = (const float*)d_in[0];
    const int*   edge    = (const int*)  d_in[1];   // [2][E]: row0 = src, row1 = dst
    const float* W_in    = (const float*)d_in[2];
    const float* b_in    = (const float*)d_in[3];
    const float* W_convs = (const float*)d_in[4];   // [3][128][128]
    const float* b_convs = (const float*)d_in[5];   // [3][128]
    const float* W_out   = (const float*)d_in[6];   // [64][512]
    const float* b_out   = (const float*)d_in[7];   // [64]

    const int N = in_sizes[0] / N_FEAT;
    const int E = in_sizes[1] / 2;

    const long plane = (long)N * N_FEAT;
    float* H   = (float*)d_ws;          // [4][N][128] hidden-state planes
    float* agg = H + 4 * plane;         // [N][128]

    const int gemmBlocks = (N + 63) / 64;
    const int zeroBlocks = (int)((plane / 4 + 255) / 256);
    const int scatBlocks = (E * 32 + 255) / 256;

    // 1) h0 = x @ W_in^T + b_in
    gin_gemm128<<<gemmBlocks, 128, 0, stream>>>(x, nullptr, W_in, b_in, H, N);

    // 2) three GIN layers: agg = scatter_add(h[src] -> dst); h = (h+agg) @ W^T + b
    for (int l = 0; l < 3; ++l) {
        gin_zero<<<zeroBlocks, 256, 0, stream>>>(agg, plane);
        gin_scatter<<<scatBlocks, 256, 0, stream>>>(H + l * plane, edge, edge + E, agg, E);
        gin_gemm128<<<gemmBlocks, 128, 0, stream>>>(
            H + l * plane, agg,
            W_convs + (long)l * N_FEAT * N_FEAT, b_convs + l * N_FEAT,
            H + (l + 1) * plane, N);
    }

    // 3) out = softmax( cat(H0..H3) @ W_out^T + b_out )
    gin_out_softmax<<<(N + 31) / 32, 64, 0, stream>>>(
        H, W_out, b_out, (float*)d_out, N, plane);
}